// PointerGenerator_9259949490200
// MI455X (gfx1250) — compile-verified
//
#include <hip/hip_runtime.h>
#include <hip/hip_bf16.h>
#include <math.h>

// Problem constants (fixed by the reference).
constexpr int B = 16;
constexpr int T = 1024;
constexpr int S = 1024;
constexpr int D = 1024;   // TARGET_SIZE == SOURCE_SIZE == 1024

typedef __attribute__((ext_vector_type(2))) float v2f;
typedef __attribute__((ext_vector_type(8))) float v8f;

// ---------------------------------------------------------------------------
// Kernel 1: gs[b,s] = dot(graph_state[b,s,:], w_s) via V_WMMA_F32_16X16X4_F32.
// One wave handles a 16-row tile (rows = flattened b*S+s). Per k-step of 4:
//   A (16x4 f32): lane l<16 -> row l, K=(k,k+1); lane l>=16 -> row l-16, K=(k+2,k+3)
//   B (4x16 f32): broadcast of w_s[k..k+3] across all 16 columns
// D accumulates; column 0 of D = the 16 dot products (lane 0 rows 0-7, lane 16 rows 8-15).
// ---------------------------------------------------------------------------
__global__ void __launch_bounds__(256)
gs_wmma_kernel(const float* __restrict__ graph, const float* __restrict__ w,
               float* __restrict__ gs) {
    const int lane = threadIdx.x & 31;
    const int wave = threadIdx.x >> 5;
    const int tile = blockIdx.x * 8 + wave;      // 0 .. (B*S/16 - 1) = 0..1023
    const int r0   = tile << 4;                  // base flattened row
    const int mrow = lane & 15;                  // which of the 16 rows this lane feeds
    const int koff = (lane >> 4) << 1;           // 0 for lanes 0-15, 2 for lanes 16-31

    const float* __restrict__ arow = graph + (size_t)(r0 + mrow) * D + koff;
    const float* __restrict__ wsrc = w + D + koff;   // w_s = w[TARGET_SIZE:]

    v8f acc = {0.f, 0.f, 0.f, 0.f, 0.f, 0.f, 0.f, 0.f};

#pragma unroll 8
    for (int k = 0; k < D; k += 4) {
        v2f a  = *(const v2f*)(arow + k);   // rows x (K,K+1|K+2,K+3)
        v2f bv = *(const v2f*)(wsrc + k);   // broadcast w pair per half-wave
        acc = __builtin_amdgcn_wmma_f32_16x16x4_f32(
            /*neg_a=*/false, a, /*neg_b=*/false, bv,
            /*c_mod=*/(short)0, acc, /*reuse_a=*/false, /*reuse_b=*/false);
    }

    // Column N=0 of D: lane 0 holds M=0..7 in acc[0..7], lane 16 holds M=8..15.
    float* __restrict__ grow = gs + r0;
    if (lane == 0) {
#pragma unroll
        for (int j = 0; j < 8; ++j) grow[j] = acc[j];
    } else if (lane == 16) {
#pragma unroll
        for (int j = 0; j < 8; ++j) grow[8 + j] = acc[j];
    }
}

// ---------------------------------------------------------------------------
// Kernel 2: p[b,:] = softmax(gs[b,:]) (stable, matches jax.nn.softmax).
// One 256-thread block per batch; 4 elements per thread.
// ---------------------------------------------------------------------------
__global__ void __launch_bounds__(256)
softmax_kernel(const float* __restrict__ gs, float* __restrict__ p) {
    __shared__ float red[256];
    const int b   = blockIdx.x;
    const int tid = threadIdx.x;
    const float* __restrict__ g = gs + b * S;

    float x[4], e[4];
    float m = -INFINITY;
#pragma unroll
    for (int i = 0; i < 4; ++i) { x[i] = g[tid + i * 256]; m = fmaxf(m, x[i]); }

    red[tid] = m;
    __syncthreads();
    for (int off = 128; off > 0; off >>= 1) {
        if (tid < off) red[tid] = fmaxf(red[tid], red[tid + off]);
        __syncthreads();
    }
    const float mx = red[0];
    __syncthreads();

    float s = 0.f;
#pragma unroll
    for (int i = 0; i < 4; ++i) { e[i] = expf(x[i] - mx); s += e[i]; }

    red[tid] = s;
    __syncthreads();
    for (int off = 128; off > 0; off >>= 1) {
        if (tid < off) red[tid] += red[tid + off];
        __syncthreads();
    }
    const float inv = 1.0f / red[0];

    float* __restrict__ pb = p + b * S;
#pragma unroll
    for (int i = 0; i < 4; ++i) pb[tid + i * 256] = e[i] * inv;
}

// ---------------------------------------------------------------------------
// Kernel 3: out[b,t,s] = p[b,s]*keep / (sum_s p[b,s]*keep + 1e-12), keep = !mask.
// One wave per (b,t) row; 8 waves (rows) per block; p[b,:] cached in LDS.
// Mask row read as 32 dwords/lane-group, masked products kept in 8 float4 regs,
// norm via wave32 shfl_xor reduction, output stored as float4.
// ---------------------------------------------------------------------------
__global__ void __launch_bounds__(256)
out_kernel(const unsigned char* __restrict__ mask, const float* __restrict__ p,
           float* __restrict__ out) {
    __shared__ float pls[S];
    const int tid  = threadIdx.x;
    const int lane = tid & 31;
    const int wave = tid >> 5;
    const int b    = blockIdx.x >> 7;            // 128 blocks per batch
    const int t    = (blockIdx.x & 127) * 8 + wave;

    const float* __restrict__ pb = p + b * S;
    for (int i = tid; i < S; i += 256) pls[i] = pb[i];
    __syncthreads();

    const size_t rowoff = ((size_t)b * T + t) * S;
    const unsigned int* __restrict__ mrow = (const unsigned int*)(mask + rowoff);

    float4 v[8];
    float norm = 0.f;
#pragma unroll
    for (int i = 0; i < 8; ++i) {
        const int s4 = i * 32 + lane;            // dword index within the row
        const unsigned int m4 = mrow[s4];
        const int s = s4 * 4;
        const float v0 = ( m4        & 0xFFu) ? 0.f : pls[s + 0];
        const float v1 = ((m4 >> 8)  & 0xFFu) ? 0.f : pls[s + 1];
        const float v2 = ((m4 >> 16) & 0xFFu) ? 0.f : pls[s + 2];
        const float v3 = ( m4 >> 24         ) ? 0.f : pls[s + 3];
        v[i] = make_float4(v0, v1, v2, v3);
        norm += (v0 + v1) + (v2 + v3);
    }

#pragma unroll
    for (int off = 16; off > 0; off >>= 1) norm += __shfl_xor(norm, off, 32);

    const float inv = 1.0f / (norm + 1e-12f);
    float4* __restrict__ orow = (float4*)(out + rowoff);
#pragma unroll
    for (int i = 0; i < 8; ++i) {
        float4 o = v[i];
        o.x *= inv; o.y *= inv; o.z *= inv; o.w *= inv;
        orow[i * 32 + lane] = o;
    }
}

// ---------------------------------------------------------------------------
// Launch. Inputs: [0]=sentence_state (UNUSED: softmax shift-invariance cancels
// it exactly), [1]=graph_state f32, [2]=mask bool(1B), [3]=w f32[2048].
// Workspace: gs (16K f32) then p (16K f32) = 128 KB.
// ---------------------------------------------------------------------------
extern "C" void kernel_launch(void* const* d_in, const int* in_sizes, int n_in,
                              void* d_out, int out_size, void* d_ws, size_t ws_size,
                              hipStream_t stream) {
    const float*         graph = (const float*)d_in[1];
    const unsigned char* mask  = (const unsigned char*)d_in[2];
    const float*         w     = (const float*)d_in[3];
    float*               out   = (float*)d_out;

    float* gs = (float*)d_ws;
    float* p  = gs + B * S;

    gs_wmma_kernel<<<dim3((B * S / 16) / 8), dim3(256), 0, stream>>>(graph, w, gs);
    softmax_kernel<<<dim3(B), dim3(256), 0, stream>>>(gs, p);
    out_kernel<<<dim3(B * T / 8), dim3(256), 0, stream>>>(mask, p, out);
}